// Mask2FormerLoss_19542101196893
// MI455X (gfx1250) — compile-verified
//
#include <hip/hip_runtime.h>
#include <hip/hip_bf16.h>
#include <math.h>

// Mask2Former loss for MI455X (gfx1250, wave32).
// B=8, Q=100, C=151, H=W=256. pred_masks (210MB) dominates -> stream once at HBM rate.
#define B_   8
#define Q_   100
#define C_   151
#define HW_  65536
#define NOOBJ (C_ - 1)        // 150
#define SPLITS 8
#define PIX_PER_SPLIT (HW_ / SPLITS)   // 8192 pixels per block

typedef __attribute__((ext_vector_type(2))) float v2f;
typedef __attribute__((ext_vector_type(8))) float v8f;

__device__ __forceinline__ float wave_sum(float v) {
#pragma unroll
  for (int off = 16; off >= 1; off >>= 1) v += __shfl_xor(v, off, 32);
  return v;
}
__device__ __forceinline__ float wave_max(float v) {
#pragma unroll
  for (int off = 16; off >= 1; off >>= 1) v = fmaxf(v, __shfl_xor(v, off, 32));
  return v;
}

// ---------------- Kernel 1: per-batch class presence -> targets/valid ----------------
__global__ void prep_kernel(const int* __restrict__ tgt,
                            int* __restrict__ ct, int* __restrict__ valid) {
  __shared__ int flags[C_];
  const int b = blockIdx.x, tid = threadIdx.x;
  if (tid < C_) flags[tid] = 0;
  __syncthreads();
  const int* g = tgt + (size_t)b * HW_;
#pragma unroll 4
  for (int i = 0; i < HW_ / 256; ++i) {
    int v = g[i * 256 + tid];
    v = v < 0 ? 0 : (v > NOOBJ ? NOOBJ : v);   // clip like reference
    flags[v] = 1;                               // benign race, all write 1
  }
  __syncthreads();
  if (tid == 0) {
    int n_all = 0;
    for (int c = 0; c < C_; ++c) n_all += flags[c];
    if (flags[0] && n_all > 1) flags[0] = 0;    // drop leading 0 iff >1 unique
    int k = 0;
    int* ctb = ct + b * Q_;
    for (int c = 0; c < C_ && k < Q_; ++c)
      if (flags[c]) ctb[k++] = c;               // present classes, ascending
    for (; k < Q_; ++k) ctb[k] = NOOBJ;         // unassigned -> no-object
    int* vb = valid + b * Q_;
    for (int q = 0; q < Q_; ++q) vb[q] = (ctb[q] != NOOBJ) ? 1 : 0;
  }
}

// ---------------- Kernel 2: CE per (b,q) row, one wave32 per row ----------------
__global__ void class_kernel(const float* __restrict__ logits,
                             const int* __restrict__ ct, float* __restrict__ ce) {
  const int tid = threadIdx.x, ln = tid & 31;
  const int r = blockIdx.x * 8 + (tid >> 5);    // 100 blocks * 8 waves = 800 rows
  const float* row = logits + (size_t)r * C_;
  float mx = -INFINITY;
  for (int j = ln; j < C_; j += 32) mx = fmaxf(mx, row[j]);
  mx = wave_max(mx);
  float s = 0.f;
  for (int j = ln; j < C_; j += 32) s += __expf(row[j] - mx);
  s = wave_sum(s);
  if (ln == 0) ce[r] = mx + __logf(s) - row[ct[r]];
}

// ---------------- Kernel 3: stream pred_masks, 4 reductions per (b,q,split) ----------
__device__ __forceinline__ void px(float x, int g, int tc,
                                   float& sb, float& si, float& sp, float& st) {
  const float t = (g == tc) ? 1.f : 0.f;
  const float e = __expf(-fabsf(x));                 // v_exp_f32
  const float r = __builtin_amdgcn_rcpf(1.f + e);    // raw v_rcp_f32, ~1ulp fine for a loss
  const float l = __logf(1.f + e);                   // v_log_f32
  const float p = (x >= 0.f) ? r : (1.f - r);        // sigmoid(x)
  sb += fmaxf(x, 0.f) - x * t + l;                   // stable BCE-with-logits
  si += p * t;
  sp += p;
  st += t;
}

__global__ void __launch_bounds__(256) mask_kernel(const float* __restrict__ masks,
                                                   const int* __restrict__ tgt,
                                                   const int* __restrict__ ct,
                                                   float* __restrict__ part) {
  __shared__ float wsum[8][4];
  const int tid = threadIdx.x;
  const int bq = blockIdx.x / SPLITS;
  const int s  = blockIdx.x - bq * SPLITS;
  const int b  = bq / Q_;
  const int tc = ct[bq];                              // uniform per block

  const float4* mp = (const float4*)(masks + (size_t)bq * HW_ + (size_t)s * PIX_PER_SPLIT);
  const int4*   gp = (const int4*)(tgt + (size_t)b * HW_ + (size_t)s * PIX_PER_SPLIT);

  float sb = 0.f, si = 0.f, sp = 0.f, st = 0.f;
#pragma unroll
  for (int it = 0; it < PIX_PER_SPLIT / 1024; ++it) { // 8 iters, coalesced b128 loads
    const int idx = it * 256 + tid;
    const float4 x4 = mp[idx];
    const int4   g4 = gp[idx];
    px(x4.x, g4.x, tc, sb, si, sp, st);
    px(x4.y, g4.y, tc, sb, si, sp, st);
    px(x4.z, g4.z, tc, sb, si, sp, st);
    px(x4.w, g4.w, tc, sb, si, sp, st);
  }

  // wave32 reduction of the 4 accumulators
  sb = wave_sum(sb); si = wave_sum(si); sp = wave_sum(sp); st = wave_sum(st);
  const int wv = tid >> 5, ln = tid & 31;
  if (ln == 0) { wsum[wv][0] = sb; wsum[wv][1] = si; wsum[wv][2] = sp; wsum[wv][3] = st; }
  __syncthreads();

  // Block-level reduction with V_WMMA_F32_16X16X4_F32:
  // A (16x4 f32): row m = quantity (m<4), K = folded wave-pairs. Documented layout:
  //   lanes 0-15:  VGPR0=K0, VGPR1=K1 for row M=lane
  //   lanes 16-31: VGPR0=K2, VGPR1=K3 for row M=lane-16
  // B = all-ones (constant => layout-invariant). D[m,n] = sum_k A[m,k].
  const int m = ln & 15;
  float a0 = 0.f, a1 = 0.f;
  if (m < 4) {
    const int kb = (ln < 16) ? 0 : 2;
    a0 = wsum[kb][m]     + wsum[kb + 4][m];
    a1 = wsum[kb + 1][m] + wsum[kb + 5][m];
  }
  v2f A;  A.x = a0;  A.y = a1;
  v2f Bo; Bo.x = 1.f; Bo.y = 1.f;
  v8f Cz = {0.f, 0.f, 0.f, 0.f, 0.f, 0.f, 0.f, 0.f};
  // all waves execute with full EXEC (no divergence around the WMMA)
  v8f D = __builtin_amdgcn_wmma_f32_16x16x4_f32(false, A, false, Bo, (short)0, Cz,
                                                false, false);
  if (tid == 0) {
    // D VGPR v, lane 0 => D[M=v, N=0]; rows 0..3 hold the four block sums
    float* o = part + (size_t)bq * (SPLITS * 4) + s * 4;
    o[0] = D[0]; o[1] = D[1]; o[2] = D[2]; o[3] = D[3];
  }
}

// ---------------- Kernel 4: fold splits -> per-query bce mean + dice ----------------
__global__ void reduce_kernel(const float* __restrict__ part, float* __restrict__ bd) {
  const int r = blockIdx.x * 256 + threadIdx.x;
  if (r >= B_ * Q_) return;
  float sb = 0.f, si = 0.f, sp = 0.f, st = 0.f;
  const float* p = part + (size_t)r * (SPLITS * 4);
#pragma unroll
  for (int s = 0; s < SPLITS; ++s) {
    sb += p[s * 4 + 0]; si += p[s * 4 + 1]; sp += p[s * 4 + 2]; st += p[s * 4 + 3];
  }
  bd[r * 2 + 0] = sb * (1.f / (float)HW_);                       // bce_q
  bd[r * 2 + 1] = 1.f - (2.f * si + 1.f) / (sp + st + 1.f);      // dice_q (SMOOTH=1)
}

// ---------------- Kernel 5: final combine ----------------
__global__ void final_kernel(const float* __restrict__ ce, const float* __restrict__ bd,
                             const int* __restrict__ valid, float* __restrict__ out) {
  __shared__ float red[256];
  __shared__ float mB[8], dB[8];
  const int tid = threadIdx.x;
  float cs = 0.f;
  for (int r = tid; r < B_ * Q_; r += 256) cs += ce[r];
  red[tid] = cs;
  __syncthreads();
  for (int o = 128; o >= 1; o >>= 1) {
    if (tid < o) red[tid] += red[tid + o];
    __syncthreads();
  }
  // wave w handles batch w (8 waves)
  const int w = tid >> 5, ln = tid & 31;
  float nv = 0.f, mb = 0.f, db = 0.f;
  for (int q = ln; q < Q_; q += 32) {
    const int r = w * Q_ + q;
    const float v = (float)valid[r];
    nv += v; mb += v * bd[r * 2 + 0]; db += v * bd[r * 2 + 1];
  }
  nv = wave_sum(nv); mb = wave_sum(mb); db = wave_sum(db);
  if (ln == 0) {
    mB[w] = (nv > 0.f) ? mb / nv : 0.f;
    dB[w] = (nv > 0.f) ? db / nv : 0.f;
  }
  __syncthreads();
  if (tid == 0) {
    const float cl = red[0] / (float)(B_ * Q_);
    float ms = 0.f, ds = 0.f;
    for (int b = 0; b < B_; ++b) { ms += mB[b]; ds += dB[b]; }
    const float ml = (ms > 0.f) ? ms / (float)B_ : 0.f;
    const float dl = (ds > 0.f) ? ds / (float)B_ : 0.f;
    out[0] = 2.f * cl + 5.f * ml + 5.f * dl;
    out[1] = cl; out[2] = ml; out[3] = dl;
  }
}

extern "C" void kernel_launch(void* const* d_in, const int* in_sizes, int n_in,
                              void* d_out, int out_size, void* d_ws, size_t ws_size,
                              hipStream_t stream) {
  const float* logits = (const float*)d_in[0];   // [8,100,151]
  const float* masks  = (const float*)d_in[1];   // [8,100,256,256]
  const int*   tgt    = (const int*)d_in[2];     // [8,256,256]
  float* out = (float*)d_out;                    // (total, class, mask, dice)

  char* ws = (char*)d_ws;                        // ~134 KB used, all slots written before read
  int*   ws_ct    = (int*)(ws);                  // 800 ints
  int*   ws_valid = (int*)(ws + 4 * 1024);       // 800 ints
  float* ws_ce    = (float*)(ws + 8 * 1024);     // 800 f
  float* ws_part  = (float*)(ws + 12 * 1024);    // 800*8*4 f = 100 KB
  float* ws_bd    = (float*)(ws + 128 * 1024);   // 800*2 f

  prep_kernel  <<<B_, 256, 0, stream>>>(tgt, ws_ct, ws_valid);
  class_kernel <<<(B_ * Q_) / 8, 256, 0, stream>>>(logits, ws_ct, ws_ce);
  mask_kernel  <<<B_ * Q_ * SPLITS, 256, 0, stream>>>(masks, tgt, ws_ct, ws_part);
  reduce_kernel<<<(B_ * Q_ + 255) / 256, 256, 0, stream>>>(ws_part, ws_bd);
  final_kernel <<<1, 256, 0, stream>>>(ws_ce, ws_bd, ws_valid, out);
}